// SimpleBiquadEQ_76141180223823
// MI455X (gfx1250) — compile-verified
//
#include <hip/hip_runtime.h>
#include <hip/hip_bf16.h>

// ---------------------------------------------------------------------------
// Cascade of 10 RBJ peaking biquads over audio[32,2,262144] (f32).
// Memory-bound workload (~134 MB min traffic, ~6 us floor at 23.3 TB/s; the
// whole tensor is L2-resident at 192 MB). The order-2 IIR recurrence is
// restructured into 16-sample chunks, where each chunk is an affine map:
//   y = K*x + t1*x[-1] + t2*x[-2] + h1*y[-1] + h2*y[-2]
// with K = 16x16 lower-triangular Toeplitz (impulse response).
//  Kernel Z : zero-y-state responses for 16 chunks at once via
//             Y(16x16) = K(16x16) @ X(16x16)  ->  4x V_WMMA_F32_16X16X4_F32,
//             plus rank-2 x-tail fixup; emits per-chunk boundary pair d_j.
//  Kernel S : per-row scan of s_{j+1} = A16*s_j + d_j (A16 = 2x2, uniform)
//             -> Hillis-Steele with per-step matrix = A16^(16*2^k).
//  Kernel C : y = y_zs + h1[m]*s1 + h2[m]*s2  (coalesced, in-place).
// Ping-pong W1/W2 across bands; band 9 correction writes d_out.
// Workspace assumption: ws_size >= ~145 MB (2*64MiB + 2*8MiB + tables).
// ---------------------------------------------------------------------------

typedef __attribute__((ext_vector_type(2))) float v2f;
typedef __attribute__((ext_vector_type(8))) float v8f;

constexpr int   B_   = 32, C_ = 2, NB_ = 10;
constexpr int   S_   = 262144;
constexpr int   ROWS = B_ * C_;               // 64
constexpr int   CHUNKS_PER_ROW = S_ / 16;     // 16384
constexpr int   TILE = 256;                   // 16 chunks per wave-tile
constexpr int   TILES_PER_ROW = S_ / TILE;    // 1024
constexpr int   FILT_STRIDE = 80;             // floats per (b,band): k,t1,t2,h1,h2

// ---------------------------------------------------------------------------
// Precompute: RBJ coefficients + 5 length-16 response sequences per (b,band).
// ---------------------------------------------------------------------------
__global__ void eq_precompute(const float* __restrict__ cf,
                              const float* __restrict__ gn,
                              const float* __restrict__ qf,
                              float* __restrict__ filt)
{
    int i = blockIdx.x * blockDim.x + threadIdx.x;
    if (i >= B_ * NB_) return;
    float f = cf[i], g = gn[i], q = qf[i];
    float w0    = 2.0f * 3.14159265358979323846f * f / 44100.0f;
    float alpha = __sinf(w0) / (2.0f * q);
    float Aamp  = __powf(10.0f, g / 40.0f);
    float cw    = __cosf(w0);
    float a0    = 1.0f + alpha / Aamp;
    float b0 = (1.0f + alpha * Aamp) / a0;
    float b1 = (-2.0f * cw) / a0;
    float b2 = (1.0f - alpha * Aamp) / a0;
    float a1 = (-2.0f * cw) / a0;
    float a2 = (1.0f - alpha / Aamp) / a0;

    float* F = filt + (size_t)i * FILT_STRIDE;
    // y[n] = inj[n] - a1*y[n-1] - a2*y[n-2], per-sequence injection/state:
    float inj[5][3] = { {b0, b1, b2},    // k  : impulse response
                        {b1, b2, 0.f},   // t1 : response to x[-1]=1
                        {b2, 0.f, 0.f},  // t2 : response to x[-2]=1
                        {0.f,0.f,0.f},   // h1 : zero-input, y[-1]=1
                        {0.f,0.f,0.f} }; // h2 : zero-input, y[-2]=1
    float is1[5] = {0.f,0.f,0.f,1.f,0.f};
    float is2[5] = {0.f,0.f,0.f,0.f,1.f};
    for (int s = 0; s < 5; ++s) {
        float ym1 = is1[s], ym2 = is2[s];
        for (int n = 0; n < 16; ++n) {
            float d = (n < 3) ? inj[s][n] : 0.f;
            float y = d - a1 * ym1 - a2 * ym2;
            F[s * 16 + n] = y;
            ym2 = ym1; ym1 = y;
        }
    }
}

// ---------------------------------------------------------------------------
// Kernel Z: zero-y-state chunk responses via f32 WMMA.
// Grid: (16, ROWS) x 256 threads (8 waves). Each wave owns strided tiles.
// ---------------------------------------------------------------------------
__global__ __launch_bounds__(256) void eq_zerostate(
    const float* __restrict__ xin, float* __restrict__ yzs,
    float2* __restrict__ dbuf, const float* __restrict__ filt, int band)
{
    const int wave = threadIdx.x >> 5;
    const int lane = threadIdx.x & 31;
    const int hi   = lane >> 4;      // lane-half selects K/M sub-blocks
    const int lo   = lane & 15;
    const int row  = blockIdx.y;
    const int b    = row / C_;
    const float* F = filt + (size_t)(b * NB_ + band) * FILT_STRIDE;
    const float* K = F;              // impulse response k[0..15]

    // A-matrix panels (16x4 f32 each): A[M][kk] = k[M-kk] (lower-tri Toeplitz).
    // ISA layout: lanes 0-15 -> M=lane, VGPR0=K0,VGPR1=K1; lanes 16-31 -> K2,K3.
    v2f a[4];
#pragma unroll
    for (int p = 0; p < 4; ++p) {
        int kk = p * 4 + 2 * hi;
        int i0 = lo - kk, i1 = lo - kk - 1;
        float x0 = (i0 >= 0) ? K[i0] : 0.f;
        float x1 = (i1 >= 0) ? K[i1] : 0.f;
        a[p].x = x0; a[p].y = x1;
    }
    // x-tail response rows for this lane's M values (uniform scalar loads + select)
    float t1m[8], t2m[8];
#pragma unroll
    for (int r = 0; r < 8; ++r) {
        t1m[r] = hi ? F[16 + r + 8] : F[16 + r];
        t2m[r] = hi ? F[32 + r + 8] : F[32 + r];
    }

    const float* xrow = xin + (size_t)row * S_;
    float*       yrow = yzs + (size_t)row * S_;
    float2*      drow = dbuf + (size_t)row * CHUNKS_PER_ROW;

    const int wavesPerRow = gridDim.x * (blockDim.x >> 5);   // 128
    for (int tile = blockIdx.x * (blockDim.x >> 5) + wave;
         tile < TILES_PER_ROW; tile += wavesPerRow) {
        const float* xt = xrow + (size_t)tile * TILE;
        __builtin_prefetch(xrow + (size_t)(tile + wavesPerRow) * TILE, 0, 3);

        // Y(16x16) = K @ X : B panel rows = in-chunk sample kk, cols = chunk (N=lo)
        v8f c = {};
#pragma unroll
        for (int p = 0; p < 4; ++p) {
            int koff = p * 4 + 2 * hi;
            v2f bm = *(const v2f*)(xt + lo * 16 + koff);
            c = __builtin_amdgcn_wmma_f32_16x16x4_f32(
                    false, a[p], false, bm, (short)0, c, false, false);
        }

        // rank-2 fixup: contributions of x[-1], x[-2] of each chunk (known
        // input). s0 is a multiple of 16, so only chunk 0 of the row lacks
        // history: do one unconditional aligned b64 load at max(s0-2,0) and
        // zero the contribution with a 0/1 weight (no exec divergence).
        int   cg  = tile * 16 + lo;        // this lane's global chunk index
        int   s0  = cg * 16;               // chunk start sample in row
        int   ti  = (cg > 0) ? (s0 - 2) : 0;
        float w   = (cg > 0) ? 1.0f : 0.0f;
        v2f xm = *(const v2f*)(xrow + ti);  // {x[-2], x[-1]}
        float xm2 = xm.x * w;
        float xm1 = xm.y * w;
#pragma unroll
        for (int r = 0; r < 8; ++r)
            c[r] += t1m[r] * xm1 + t2m[r] * xm2;

        // store zero-state tile: lane covers samples s0 + hi*8 + (0..7)
        float* dst = yrow + (size_t)s0 + hi * 8;
        *(float4*)(dst)     = make_float4(c[0], c[1], c[2], c[3]);
        *(float4*)(dst + 4) = make_float4(c[4], c[5], c[6], c[7]);

        // chunk boundary pair d_j = (y_zs[15], y_zs[14]) lives in hi lanes r=7,6
        if (hi) drow[cg] = make_float2(c[7], c[6]);
    }
}

// ---------------------------------------------------------------------------
// Kernel S: per-row affine scan  s_{j+1} = A16 * s_j + d_j  over 16384 chunks.
// One 1024-thread workgroup per row; 16 chunks serial per thread, then a
// Hillis-Steele scan whose per-step matrix is A16^(16*2^k) (uniform squarings).
// ---------------------------------------------------------------------------
__global__ __launch_bounds__(1024) void eq_scan(
    const float2* __restrict__ dbuf, float2* __restrict__ sbuf,
    const float* __restrict__ filt, int band)
{
    __shared__ float2 sv[1024];
    const int row = blockIdx.x;
    const int t   = threadIdx.x;
    const int b   = row / C_;
    const float* F = filt + (size_t)(b * NB_ + band) * FILT_STRIDE;
    // A16 maps incoming (y[-1],y[-2]) to outgoing (y[15],y[14]) under zero input
    const float m00 = F[48 + 15], m01 = F[64 + 15];
    const float m10 = F[48 + 14], m11 = F[64 + 14];

    const float2* drow = dbuf + (size_t)row * CHUNKS_PER_ROW;
    float2*       srow = sbuf + (size_t)row * CHUNKS_PER_ROW;

    float2 d[16];
    float ax = 0.f, ay = 0.f;                 // thread-local affine aggregate (vector part)
#pragma unroll
    for (int i = 0; i < 16; ++i) {
        d[i] = drow[t * 16 + i];
        float nx = m00 * ax + m01 * ay + d[i].x;
        float ny = m10 * ax + m11 * ay + d[i].y;
        ax = nx; ay = ny;
    }
    sv[t] = make_float2(ax, ay);

    // P = A16^16 via 4 squarings
    float p00 = m00, p01 = m01, p10 = m10, p11 = m11;
#pragma unroll
    for (int s = 0; s < 4; ++s) {
        float q00 = p00*p00 + p01*p10, q01 = p00*p01 + p01*p11;
        float q10 = p10*p00 + p11*p10, q11 = p10*p01 + p11*p11;
        p00 = q00; p01 = q01; p10 = q10; p11 = q11;
    }
    __syncthreads();

    for (int off = 1; off < 1024; off <<= 1) {
        float2 prev = make_float2(0.f, 0.f);
        if (t >= off) prev = sv[t - off];
        __syncthreads();
        if (t >= off) {
            float2 cur = sv[t];
            cur.x += p00 * prev.x + p01 * prev.y;
            cur.y += p10 * prev.x + p11 * prev.y;
            sv[t] = cur;
        }
        __syncthreads();
        float q00 = p00*p00 + p01*p10, q01 = p00*p01 + p01*p11;
        float q10 = p10*p00 + p11*p10, q11 = p10*p01 + p11*p11;
        p00 = q00; p01 = q01; p10 = q10; p11 = q11;
    }

    // replay: emit incoming state for each of this thread's chunks
    float sx = 0.f, sy = 0.f;
    if (t > 0) { float2 pr = sv[t - 1]; sx = pr.x; sy = pr.y; }
#pragma unroll
    for (int i = 0; i < 16; ++i) {
        srow[t * 16 + i] = make_float2(sx, sy);
        float nx = m00 * sx + m01 * sy + d[i].x;
        float ny = m10 * sx + m11 * sy + d[i].y;
        sx = nx; sy = ny;
    }
}

// ---------------------------------------------------------------------------
// Kernel C: y[n] = y_zs[n] + h1[n%16]*s1 + h2[n%16]*s2  (in-place capable)
// ---------------------------------------------------------------------------
__global__ __launch_bounds__(256) void eq_correct(
    const float* __restrict__ yzs, const float2* __restrict__ sbuf,
    const float* __restrict__ filt, float* __restrict__ out, int band)
{
    size_t n = (size_t)blockIdx.x * blockDim.x + threadIdx.x;
    const size_t total = (size_t)ROWS * S_;
    if (n >= total) return;
    int row = (int)(n / S_);
    int off = (int)(n % S_);
    int j = off >> 4, m = off & 15;
    int b = row / C_;
    const float* F = filt + (size_t)(b * NB_ + band) * FILT_STRIDE;
    float2 s = sbuf[(size_t)row * CHUNKS_PER_ROW + j];
    out[n] = yzs[n] + F[48 + m] * s.x + F[64 + m] * s.y;
}

// ---------------------------------------------------------------------------
extern "C" void kernel_launch(void* const* d_in, const int* in_sizes, int n_in,
                              void* d_out, int out_size, void* d_ws, size_t ws_size,
                              hipStream_t stream)
{
    const float* audio = (const float*)d_in[0];
    const float* cf    = (const float*)d_in[1];
    const float* gn    = (const float*)d_in[2];
    const float* qf    = (const float*)d_in[3];
    float* out = (float*)d_out;

    const size_t Nf = (size_t)ROWS * S_;
    char* ws = (char*)d_ws;
    float*  W1   = (float*)ws;   ws += Nf * sizeof(float);
    float*  W2   = (float*)ws;   ws += Nf * sizeof(float);
    float2* Dbuf = (float2*)ws;  ws += (size_t)ROWS * CHUNKS_PER_ROW * sizeof(float2);
    float2* Sbuf = (float2*)ws;  ws += (size_t)ROWS * CHUNKS_PER_ROW * sizeof(float2);
    float*  Filt = (float*)ws;

    eq_precompute<<<2, 256, 0, stream>>>(cf, gn, qf, Filt);

    dim3 zgrid(16, ROWS);
    int  cgrid = (int)((Nf + 255) / 256);
    for (int band = 0; band < NB_; ++band) {
        const float* xin = (band == 0) ? audio : ((band & 1) ? W1 : W2);
        float* yz   = (band & 1) ? W2 : W1;
        float* yout = (band == NB_ - 1) ? out : yz;   // in-place except last band
        eq_zerostate<<<zgrid, 256, 0, stream>>>(xin, yz, Dbuf, Filt, band);
        eq_scan<<<ROWS, 1024, 0, stream>>>(Dbuf, Sbuf, Filt, band);
        eq_correct<<<cgrid, 256, 0, stream>>>(yz, Sbuf, Filt, yout, band);
    }
}